// HeadAttention_42640435315379
// MI455X (gfx1250) — compile-verified
//
#include <hip/hip_runtime.h>

typedef float v2f __attribute__((ext_vector_type(2)));
typedef float v8f __attribute__((ext_vector_type(8)));

#define EMB   1024
#define HEAD  64
#define BATCH 8
#define SEQ   2048

// ---------------------------------------------------------------------------
// Kernel 1: fused Q/K/V projection.  One wave (32 lanes) computes a 16x64
// output tile:  out[16,64] = x[16,1024] @ W[1024,64] + bias, using
// v_wmma_f32_16x16x4_f32 (exact f32 matrix pipe).
// grid = (B*S/16, 3): blockIdx.y selects Q / K / V.
// ---------------------------------------------------------------------------
__global__ __launch_bounds__(32) void qkv_proj_kernel(
    const float* __restrict__ x,
    const float* __restrict__ Wq, const float* __restrict__ bq,
    const float* __restrict__ Wk, const float* __restrict__ bk,
    const float* __restrict__ Wv, const float* __restrict__ bv,
    float* __restrict__ Qo, float* __restrict__ Ko, float* __restrict__ Vo)
{
    const int tile  = blockIdx.x;                 // 16-row tile over B*S
    const int which = blockIdx.y;                 // 0=Q 1=K 2=V
    const float* W    = (which == 0) ? Wq : (which == 1) ? Wk : Wv;
    const float* bias = (which == 0) ? bq : (which == 1) ? bk : bv;
    float*       out  = (which == 0) ? Qo : (which == 1) ? Ko : Vo;

    const int lane = threadIdx.x;
    const int m    = lane & 15;                   // M (A) / N (B,C) index
    const int kh   = lane >> 4;                   // K half select

    const int row0 = tile * 16;
    const float* xrow = x + (size_t)(row0 + m) * EMB + 2 * kh;

    v8f acc0 = {}, acc1 = {}, acc2 = {}, acc3 = {};

    for (int k0 = 0; k0 < EMB; k0 += 4) {
        // A 16x4: lane m holds x[row0+m][k0 + 2*kh + {0,1}]
        v2f a = *(const v2f*)(xrow + k0);
        // B 4x16: lane m holds W[k0 + 2*kh + {0,1}][n0 + m]
        const float* wr = W + (size_t)(k0 + 2 * kh) * HEAD + m;
        v2f b0; b0.x = wr[0];  b0.y = wr[HEAD];
        v2f b1; b1.x = wr[16]; b1.y = wr[HEAD + 16];
        v2f b2; b2.x = wr[32]; b2.y = wr[HEAD + 32];
        v2f b3; b3.x = wr[48]; b3.y = wr[HEAD + 48];
        acc0 = __builtin_amdgcn_wmma_f32_16x16x4_f32(false, a, false, b0, (short)0, acc0, false, false);
        acc1 = __builtin_amdgcn_wmma_f32_16x16x4_f32(false, a, false, b1, (short)0, acc1, false, false);
        acc2 = __builtin_amdgcn_wmma_f32_16x16x4_f32(false, a, false, b2, (short)0, acc2, false, false);
        acc3 = __builtin_amdgcn_wmma_f32_16x16x4_f32(false, a, false, b3, (short)0, acc3, false, false);
    }

    // C/D layout: VGPR r, lane -> element [r + 8*kh][m]
    #pragma unroll
    for (int r = 0; r < 8; ++r) {
        float* orow = out + (size_t)(row0 + r + 8 * kh) * HEAD;
        orow[m]      = acc0[r] + bias[m];
        orow[16 + m] = acc1[r] + bias[16 + m];
        orow[32 + m] = acc2[r] + bias[32 + m];
        orow[48 + m] = acc3[r] + bias[48 + m];
    }
}

// ---------------------------------------------------------------------------
// Kernel 2: fused causal flash-attention.  One wave per (batch, 16-query
// tile).  Streams key/value 16-row tiles up to the diagonal with an online
// softmax.  QK^T and P@V both use v_wmma_f32_16x16x4_f32.  The probability
// tile is rotated from C/D layout to A layout through LDS (in-order DS ops
// within a wave).
// grid = (S/16, B)
// ---------------------------------------------------------------------------
__global__ __launch_bounds__(32) void attn_kernel(
    const float* __restrict__ Q, const float* __restrict__ K,
    const float* __restrict__ V, float* __restrict__ Out)
{
    __shared__ __align__(16) float qs[16 * HEAD];  // Q tile, row-major
    __shared__ __align__(16) float ps[16 * 16];    // P tile, row-major

    const int b    = blockIdx.y;
    const int qt   = blockIdx.x;
    const int q0   = qt * 16;
    const int lane = threadIdx.x;
    const int m    = lane & 15;
    const int kh   = lane >> 4;

    const float* Qb = Q + ((size_t)b * SEQ + q0) * HEAD;
    const float* Kb = K + (size_t)b * SEQ * HEAD;
    const float* Vb = V + (size_t)b * SEQ * HEAD;

    // Stage the 16x64 Q tile into LDS (rows are contiguous in memory).
    for (int i = lane; i < 16 * HEAD / 4; i += 32)
        ((float4*)qs)[i] = ((const float4*)Qb)[i];

    float mrow[8], lrow[8];
    #pragma unroll
    for (int r = 0; r < 8; ++r) { mrow[r] = -1e30f; lrow[r] = 0.0f; }
    v8f o0 = {}, o1 = {}, o2 = {}, o3 = {};

    for (int kt = 0; kt <= qt; ++kt) {
        const int kv0 = kt * 16;

        // ---- S = Q @ K^T : 16 WMMA steps over the 64 head dims ----
        v8f s = {};
        #pragma unroll
        for (int kk = 0; kk < HEAD / 4; ++kk) {
            const int h = kk * 4 + 2 * kh;
            v2f a  = *(const v2f*)(qs + m * HEAD + h);              // Q[q0+m][h..h+1]
            v2f bf = *(const v2f*)(Kb + (size_t)(kv0 + m) * HEAD + h); // K[kv0+m][h..h+1]
            s = __builtin_amdgcn_wmma_f32_16x16x4_f32(false, a, false, bf, (short)0, s, false, false);
        }

        // ---- scale, causal mask, online softmax update ----
        const bool diag = (kt == qt);
        #pragma unroll
        for (int r = 0; r < 8; ++r) {
            const int row = r + 8 * kh;                 // local query row
            float sv = s[r] * 0.125f;                   // 1/sqrt(64)
            if (diag && (m > row)) sv = -1e30f;         // key col > query row
            // row max across the 16 lanes of this half (butterfly)
            float rm = sv;
            rm = fmaxf(rm, __shfl_xor(rm, 1, 32));
            rm = fmaxf(rm, __shfl_xor(rm, 2, 32));
            rm = fmaxf(rm, __shfl_xor(rm, 4, 32));
            rm = fmaxf(rm, __shfl_xor(rm, 8, 32));
            const float mnew  = fmaxf(mrow[r], rm);
            const float p     = __expf(sv - mnew);
            const float scale = __expf(mrow[r] - mnew);
            float psum = p;
            psum += __shfl_xor(psum, 1, 32);
            psum += __shfl_xor(psum, 2, 32);
            psum += __shfl_xor(psum, 4, 32);
            psum += __shfl_xor(psum, 8, 32);
            lrow[r] = lrow[r] * scale + psum;
            mrow[r] = mnew;
            o0[r] *= scale; o1[r] *= scale; o2[r] *= scale; o3[r] *= scale;
            ps[(r + 8 * kh) * 16 + m] = p;              // C-layout -> row-major LDS
        }

        // ---- O += P @ V : P(16x16) from LDS as A, V tile streamed as B ----
        #pragma unroll
        for (int kk = 0; kk < 4; ++kk) {
            const int j = kk * 4 + 2 * kh;              // key index within tile
            v2f a = *(const v2f*)(ps + m * 16 + j);
            const float* vp = Vb + (size_t)(kv0 + j) * HEAD + m;
            v2f b0; b0.x = vp[0];  b0.y = vp[HEAD];
            v2f b1; b1.x = vp[16]; b1.y = vp[HEAD + 16];
            v2f b2; b2.x = vp[32]; b2.y = vp[HEAD + 32];
            v2f b3; b3.x = vp[48]; b3.y = vp[HEAD + 48];
            o0 = __builtin_amdgcn_wmma_f32_16x16x4_f32(false, a, false, b0, (short)0, o0, false, false);
            o1 = __builtin_amdgcn_wmma_f32_16x16x4_f32(false, a, false, b1, (short)0, o1, false, false);
            o2 = __builtin_amdgcn_wmma_f32_16x16x4_f32(false, a, false, b2, (short)0, o2, false, false);
            o3 = __builtin_amdgcn_wmma_f32_16x16x4_f32(false, a, false, b3, (short)0, o3, false, false);
        }
    }

    // ---- epilogue: divide by row sums, write out ----
    #pragma unroll
    for (int r = 0; r < 8; ++r) {
        const float inv = 1.0f / lrow[r];
        float* orow = Out + ((size_t)b * SEQ + q0 + r + 8 * kh) * HEAD;
        orow[m]      = o0[r] * inv;
        orow[16 + m] = o1[r] * inv;
        orow[32 + m] = o2[r] * inv;
        orow[48 + m] = o3[r] * inv;
    }
}

// ---------------------------------------------------------------------------
extern "C" void kernel_launch(void* const* d_in, const int* in_sizes, int n_in,
                              void* d_out, int out_size, void* d_ws, size_t ws_size,
                              hipStream_t stream) {
    const float* x  = (const float*)d_in[0];
    const float* Wq = (const float*)d_in[1];
    const float* bq = (const float*)d_in[2];
    const float* Wk = (const float*)d_in[3];
    const float* bk = (const float*)d_in[4];
    const float* Wv = (const float*)d_in[5];
    const float* bv = (const float*)d_in[6];
    float* out = (float*)d_out;

    const size_t qkv_elems = (size_t)BATCH * SEQ * HEAD;  // 1M floats = 4MB each
    float* Qw = (float*)d_ws;
    float* Kw = Qw + qkv_elems;
    float* Vw = Kw + qkv_elems;

    dim3 g1(BATCH * SEQ / 16, 3);
    qkv_proj_kernel<<<g1, 32, 0, stream>>>(x, Wq, bq, Wk, bk, Wv, bv, Qw, Kw, Vw);

    dim3 g2(SEQ / 16, BATCH);
    attn_kernel<<<g2, 32, 0, stream>>>(Qw, Kw, Vw, out);
}